// MLPsim_47218870452949
// MI455X (gfx1250) — compile-verified
//
#include <hip/hip_runtime.h>
#include <math.h>

// Problem constants (from reference setup_inputs)
#define BB 4
#define MM 4096
#define NN 4096
#define FF 8
#define HH 128
#define OO 8

typedef float v2f __attribute__((ext_vector_type(2)));
typedef float v8f __attribute__((ext_vector_type(8)));

// D = A(16x4 f32) * B(4x16 f32) + C(16x16 f32)
#define WMMA_F32(a, b, c) \
    __builtin_amdgcn_wmma_f32_16x16x4_f32(false, (a), false, (b), (short)0, (c), false, false)

__device__ __forceinline__ float elu1(float x) {
    return x > 0.0f ? x : (__expf(x) - 1.0f);
}

__global__ __launch_bounds__(128, 1) void gnn_mlp_fused(
    const float* __restrict__ feat,  // [B, N, F]
    const float* __restrict__ adj,   // [B, M, N]
    const float* __restrict__ W1,    // [H, F]
    const float* __restrict__ b1,    // [H]
    const float* __restrict__ W2,    // [H, H]
    const float* __restrict__ b2,    // [H]
    const float* __restrict__ W3,    // [O, H]
    const float* __restrict__ b3,    // [O]
    float* __restrict__ out)         // [B, M, O]
{
    // ---- LDS layouts (pitches chosen for bank-conflict-free access) ----
    __shared__ float sW1T[FF][HH + 1];    // sW1T[k][n] = W1[n][k]
    __shared__ float sW2T[HH][HH + 1];    // sW2T[k][n] = W2[n][k]
    __shared__ float sW3T[HH][OO + 1];    // sW3T[k][n] = W3[n][k]
    __shared__ float sb1[HH], sb2[HH], sb3[OO];
    __shared__ float sBm[FF][17];         // sBm[f][m]  = (adj@feat)(m,f), reduced via ds_add_f32
    __shared__ float sH1[HH][17];         // sH1[k][m]  k-major activations
    __shared__ float sH2[HH][17];

    const int tid  = threadIdx.x;       // 0..127
    const int wave = tid >> 5;          // 0..3 (wave32)
    const int lane = tid & 31;

    const int batch = blockIdx.x / (MM / 16);
    const int m0    = (blockIdx.x % (MM / 16)) * 16;

    // ---- Cooperative weight load + transpose into LDS (coalesced global reads) ----
    for (int i = tid; i < HH * HH; i += 128) {
        int n = i >> 7, k = i & 127;
        sW2T[k][n] = W2[i];
    }
    for (int i = tid; i < HH * FF; i += 128) {
        int n = i >> 3, k = i & 7;
        sW1T[k][n] = W1[i];
    }
    for (int i = tid; i < OO * HH; i += 128) {
        int n = i >> 7, k = i & 127;
        sW3T[k][n] = W3[i];
    }
    if (tid < HH) { sb1[tid] = b1[tid]; sb2[tid] = b2[tid]; }
    if (tid < OO) { sb3[tid] = b3[tid]; }
    for (int i = tid; i < FF * 17; i += 128) ((float*)sBm)[i] = 0.0f;
    __syncthreads();

    // WMMA lane decomposition (ISA 7.12.2, wave32):
    //   A 16x4:  lane&15 = M row,  VGPR0/1 = K = khalf, khalf+1  (khalf = 2*(lane>=16))
    //   B 4x16:  lane&15 = N col,  VGPR0/1 = K = khalf, khalf+1
    //   C 16x16: VGPR r -> (m = r + 8*(lane>=16), n = lane&15)
    const int mrow  = lane & 15;
    const int khalf = (lane >> 4) << 1;   // 0 or 2
    const int muppr = (lane >> 4) << 3;   // 0 or 8 (C-tile row offset)

    // ================= Phase 1: bmat = adj @ feat  (K = 4096 split over 4 waves) ===========
    {
        const float* adjRow = adj + ((size_t)batch * MM + (m0 + mrow)) * (size_t)NN;
        const float* featB  = feat + (size_t)batch * NN * FF;

        // Two independent accumulation chains to keep the matrix pipe busy.
        v8f acc0 = {};
        v8f acc1 = {};
        const int k0 = wave * (NN / 4);
        const int k1 = k0 + (NN / 4);
        #pragma unroll 4
        for (int k = k0; k < k1; k += 8) {
            if ((k & 31) == 0)  // prefetch this row's adj stream ~4KB ahead
                __builtin_prefetch(adjRow + k + 1024, 0, 0);

            // ---- step 0: columns k .. k+3 ----
            // A tile: adj[m0+mrow][k+khalf .. +1]; NT: 256MB single-use stream,
            // keep it out of the way of L2-resident feat/weights.
            v2f a0 = __builtin_nontemporal_load((const v2f*)(adjRow + k + khalf));
            const float* fp0 = featB + (size_t)(k + khalf) * FF + (mrow & 7);
            float f00 = fp0[0];
            float f01 = fp0[FF];
            v2f b0;
            b0.x = (mrow < FF) ? f00 : 0.0f;   // cndmask keeps EXEC full for WMMA
            b0.y = (mrow < FF) ? f01 : 0.0f;
            acc0 = WMMA_F32(a0, b0, acc0);

            // ---- step 1: columns k+4 .. k+7 ----
            v2f a1 = __builtin_nontemporal_load((const v2f*)(adjRow + k + 4 + khalf));
            const float* fp1 = featB + (size_t)(k + 4 + khalf) * FF + (mrow & 7);
            float f10 = fp1[0];
            float f11 = fp1[FF];
            v2f b1v;
            b1v.x = (mrow < FF) ? f10 : 0.0f;
            b1v.y = (mrow < FF) ? f11 : 0.0f;
            acc1 = WMMA_F32(a1, b1v, acc1);
        }

        // Cross-wave reduction of partial C tiles into sBm via LDS float atomics
        if (mrow < FF) {
            #pragma unroll
            for (int r = 0; r < 8; ++r) {
                atomicAdd(&sBm[mrow][r + muppr], acc0[r] + acc1[r]);   // ds_add_f32
            }
        }
    }
    __syncthreads();

    // ================= Phase 2: h1 = elu(bmat @ W1^T + b1)   [16x8 @ 8x128] ================
    #pragma unroll
    for (int t = 0; t < 2; ++t) {
        const int n0 = (wave * 2 + t) * 16;
        v8f c = {};
        #pragma unroll
        for (int kb = 0; kb < FF; kb += 4) {
            v2f a, b;
            a.x = sBm[kb + khalf][mrow];
            a.y = sBm[kb + khalf + 1][mrow];
            b.x = sW1T[kb + khalf][n0 + mrow];
            b.y = sW1T[kb + khalf + 1][n0 + mrow];
            c = WMMA_F32(a, b, c);
        }
        const int n = n0 + mrow;
        const float bias = sb1[n];
        #pragma unroll
        for (int r = 0; r < 8; ++r) {
            sH1[n][r + muppr] = elu1(c[r] + bias);
        }
    }
    __syncthreads();

    // ================= Phase 3: h2 = elu(h1 @ W2^T + b2)   [16x128 @ 128x128] ==============
    #pragma unroll
    for (int t = 0; t < 2; ++t) {
        const int n0 = (wave * 2 + t) * 16;
        v8f c = {};
        #pragma unroll 4
        for (int kb = 0; kb < HH; kb += 4) {
            v2f a, b;
            a.x = sH1[kb + khalf][mrow];
            a.y = sH1[kb + khalf + 1][mrow];
            b.x = sW2T[kb + khalf][n0 + mrow];
            b.y = sW2T[kb + khalf + 1][n0 + mrow];
            c = WMMA_F32(a, b, c);
        }
        const int n = n0 + mrow;
        const float bias = sb2[n];
        #pragma unroll
        for (int r = 0; r < 8; ++r) {
            sH2[n][r + muppr] = elu1(c[r] + bias);
        }
    }
    __syncthreads();

    // ================= Phase 4: c = h2 @ W3^T + b3   [16x128 @ 128x8] (wave 0) =============
    if (wave == 0) {
        v8f c = {};
        #pragma unroll 4
        for (int kb = 0; kb < HH; kb += 4) {
            v2f a, b;
            a.x = sH2[kb + khalf][mrow];
            a.y = sH2[kb + khalf + 1][mrow];
            float w0 = sW3T[kb + khalf][mrow & 7];
            float w1 = sW3T[kb + khalf + 1][mrow & 7];
            b.x = (mrow < OO) ? w0 : 0.0f;
            b.y = (mrow < OO) ? w1 : 0.0f;
            c = WMMA_F32(a, b, c);
        }
        if (mrow < OO) {
            const float bias = sb3[mrow];
            float* op = out + ((size_t)batch * MM + m0) * OO + mrow;
            #pragma unroll
            for (int r = 0; r < 8; ++r) {
                op[(size_t)(r + muppr) * OO] = c[r] + bias;
            }
        }
    }
}

extern "C" void kernel_launch(void* const* d_in, const int* in_sizes, int n_in,
                              void* d_out, int out_size, void* d_ws, size_t ws_size,
                              hipStream_t stream) {
    (void)in_sizes; (void)n_in; (void)out_size; (void)d_ws; (void)ws_size;
    const float* feat = (const float*)d_in[0];
    const float* adj  = (const float*)d_in[1];
    const float* W1   = (const float*)d_in[2];
    const float* b1   = (const float*)d_in[3];
    const float* W2   = (const float*)d_in[4];
    const float* b2   = (const float*)d_in[5];
    const float* W3   = (const float*)d_in[6];
    const float* b3   = (const float*)d_in[7];
    float* out = (float*)d_out;

    dim3 grid(BB * (MM / 16));   // 1024 blocks, one 16-row tile each
    dim3 block(128);             // 4 wave32 waves
    gnn_mlp_fused<<<grid, block, 0, stream>>>(feat, adj, W1, b1, W2, b2, W3, b3, out);
}